// AttentiveChildSumTreeLSTMCell_61366492725516
// MI455X (gfx1250) — compile-verified
//
#include <hip/hip_runtime.h>
#include <math.h>

// ---------------------------------------------------------------------------
// AttentiveChildSumTreeLSTMCell  (H = E = I = 2048, N = 32, A = 2048)  fp32
// Memory-bound (~181 MB fp32 weights, ~0.9 GFLOP -> ~4.9 FLOP/byte).
//  - batch-32 GEMMs: v_wmma_f32_16x16x4_f32, W tiles streamed by the
//    Tensor Data Mover (tensor_load_to_lds + s_wait_tensorcnt), double
//    buffered, D# pad feature gives the conflict-free stride-36 LDS layout.
//  - batch-1 GEMVs: one wave per row, float4 streams, wave32 shfl reduce.
// ---------------------------------------------------------------------------

typedef __attribute__((ext_vector_type(2))) float v2f;
typedef __attribute__((ext_vector_type(8))) float v8f;
typedef __attribute__((ext_vector_type(4))) unsigned int u32x4;
typedef __attribute__((ext_vector_type(8))) int i32x8;
typedef __attribute__((ext_vector_type(4))) int i32x4;

#if defined(__HIP_DEVICE_COMPILE__) && defined(__gfx1250__) && __has_builtin(__builtin_amdgcn_wmma_f32_16x16x4_f32)
#define HAVE_WMMA_F32 1
#else
#define HAVE_WMMA_F32 0
#endif

#if defined(__HIP_DEVICE_COMPILE__) && defined(__gfx1250__) && \
    __has_builtin(__builtin_amdgcn_tensor_load_to_lds) && \
    __has_builtin(__builtin_amdgcn_s_wait_tensorcnt)
#define HAVE_TDM 1
#else
#define HAVE_TDM 0
#endif

__device__ __forceinline__ float waveReduceSum(float v) {
#pragma unroll
  for (int o = 16; o > 0; o >>= 1) v += __shfl_xor(v, o, 32);
  return v;
}

__device__ __forceinline__ float sigmoidf_(float x) {
  return 1.0f / (1.0f + expf(-x));
}

#if HAVE_TDM
// TDM: DMA one W tile [128 rows x 32 f32] (row stride K elems) into LDS with
// 4-DWORD padding every 32 DWORDs  ->  LDS row stride 36 floats (bank-safe).
// D# per CDNA5 ISA 8.3/8.4: group0 = {count/flags, lds_addr, global_addr, type=2},
// group1 = {data_size=4B, pad(int=32dw,amt=4dw), tensor dims, tile 32x128, stride K}.
// This toolchain exposes the 6-arg builtin:
//   (u32x4 g0, i32x8 g1, i32x4 g2, i32x4 g3, i32x8 extra, i32 cpol)
// groups 2/3 + extra are unused for a 2-D tile -> zero-filled.
__device__ __forceinline__ void tdm_load_w_tile(const float* W, int K, int cb,
                                                int k0, float* lds_dst) {
  unsigned long long ga = (unsigned long long)(uintptr_t)W +
                          4ull * ((unsigned long long)cb * (unsigned)K + (unsigned)k0);
  unsigned int lds_addr = (unsigned int)(uintptr_t)lds_dst;  // low 32 = LDS offset
  u32x4 g0;
  g0.x = 1u;                                            // count=1 (valid), user mode
  g0.y = lds_addr;                                      // [63:32] lds_addr
  g0.z = (unsigned int)(ga & 0xFFFFFFFFull);            // [95:64] global_addr lo
  g0.w = ((unsigned int)(ga >> 32) & 0x01FFFFFFu)       // [120:96] global_addr hi
         | (2u << 30);                                  // [127:126] type=2 (image)
  i32x8 g1;
  g1[0] = (int)((2u << 16)      // data_size = 4 bytes
                | (1u << 20)    // pad_enable
                | (4u << 22)    // pad_interval: 32 DWORDs
                | (3u << 25));  // pad_amount : 4 DWORDs  -> stride 36 floats
  g1[1] = 0;                                  // tensor_dim0 (=0x100000) lo16 << 16
  g1[2] = 0x10;                               // tensor_dim0 hi16 | tensor_dim1 lo16<<16
  g1[3] = 0x10 | (32 << 16);                  // tensor_dim1 hi16 | tile_dim0 = 32
  g1[4] = 128;                                // tile_dim1 = 128 rows, tile_dim2 = 0
  g1[5] = K;                                  // tensor_dim0_stride lo32 (elems)
  g1[6] = 0;                                  // stride0 hi16 | stride1 lo16
  g1[7] = 0;
  i32x4 z4 = {0, 0, 0, 0};                    // groups 2/3 unused (2D tile)
  i32x8 z8 = {0, 0, 0, 0, 0, 0, 0, 0};
  __builtin_amdgcn_tensor_load_to_lds(g0, g1, z4, z4, z8, 0);
}
#endif

// ---------------------------------------------------------------------------
// C[32, ncols] = [Xrows | broadcast(xbroad)] (32 x (K1+K2)) @ W[ncols, K1+K2]^T
// mode 0: + bias[col] (may be null), no act.   mode 1: tanh.
// 256 threads = 8 waves; wave owns 16 output cols, both 16-row M-tiles.
// ---------------------------------------------------------------------------
__global__ __launch_bounds__(256) void gemm32_wmma_kernel(
    const float* __restrict__ Xrows, int K1,
    const float* __restrict__ xbroad, int K2,
    const float* __restrict__ W, const float* __restrict__ bias,
    float* __restrict__ C, int ncols, int mode) {
  constexpr int KT  = 32;
  constexpr int STR = 36;                 // padded LDS row stride (floats)
  __shared__ float Xs[32 * STR];
  __shared__ float Ws[2][128 * STR];

  const int tid  = threadIdx.x;
  const int lane = tid & 31;
  const int wave = tid >> 5;
  const int lm   = lane & 15;
  const int half = lane >> 4;
  const int cb   = blockIdx.x * 128;      // first output column of this block
  const int K    = K1 + K2;

  v8f acc0 = {};   // rows 0..15
  v8f acc1 = {};   // rows 16..31

#if HAVE_TDM
  if (wave == 0) tdm_load_w_tile(W, K, cb, 0, &Ws[0][0]);
#endif

  int buf = 0;
  for (int k0 = 0; k0 < K; k0 += KT, buf ^= 1) {
    // ---- stage X tile: 32 rows x 32 (8 threads/row, one float4 each) ----
    {
      const int r  = tid >> 3;
      const int c4 = (tid & 7) * 4;
      float4 v;
      if (k0 < K1) v = *(const float4*)&Xrows[(size_t)r * K1 + k0 + c4];
      else         v = *(const float4*)&xbroad[(k0 - K1) + c4];  // row-bcast
      *(float4*)&Xs[r * STR + c4] = v;
    }

    float* Wc = &Ws[buf][0];
#if HAVE_TDM
    // DMA next tile while this one finishes; TENSORcnt is in-order per wave.
    if (wave == 0) {
      if (k0 + KT < K) {
        tdm_load_w_tile(W, K, cb, k0 + KT, &Ws[buf ^ 1][0]);
        __builtin_amdgcn_s_wait_tensorcnt(1);   // current tile complete
      } else {
        __builtin_amdgcn_s_wait_tensorcnt(0);
      }
    }
#else
    // fallback: synchronous staging, 128 rows x 32 floats
    {
      int r        = tid >> 3;
      const int c4 = (tid & 7) * 4;
#pragma unroll
      for (int it = 0; it < 4; ++it, r += 32) {
        const float* src = &W[(size_t)(cb + r) * K + k0 + c4];
        *(float4*)&Wc[r * STR + c4] = *(const float4*)src;
        if (k0 + KT < K) __builtin_prefetch(src + KT, 0, 0);
      }
    }
#endif
    __syncthreads();   // W tile in LDS (TDM done + barrier), X stores visible

#if HAVE_WMMA_F32
#pragma unroll
    for (int kk = 0; kk < KT; kk += 4) {
      // A 16x4 f32: lanes 0-15 hold K=k,k+1; lanes 16-31 hold K=k+2,k+3
      const int kc = kk + 2 * half;
      v2f a0 = *(const v2f*)&Xs[lm * STR + kc];
      v2f a1 = *(const v2f*)&Xs[(16 + lm) * STR + kc];
      v2f b  = *(const v2f*)&Wc[(wave * 16 + lm) * STR + kc];
      acc0 = __builtin_amdgcn_wmma_f32_16x16x4_f32(false, a0, false, b,
                                                   (short)0, acc0, false, false);
      acc1 = __builtin_amdgcn_wmma_f32_16x16x4_f32(false, a1, false, b,
                                                   (short)0, acc1, false, false);
    }
#else
    // layout-identical VALU fallback: lane owns (m = v + 8*half (+16), n = lm)
    for (int kk = 0; kk < KT; ++kk) {
      float w = Wc[(wave * 16 + lm) * STR + kk];
#pragma unroll
      for (int v = 0; v < 8; ++v) {
        acc0[v] += Xs[(v + 8 * half) * STR + kk] * w;
        acc1[v] += Xs[(16 + v + 8 * half) * STR + kk] * w;
      }
    }
#endif
    __syncthreads();   // compute done: safe to overwrite Xs / other W buffer
  }

  // ---- epilogue: D layout (VGPR v: lanes0-15 -> M=v, lanes16-31 -> M=v+8)
  const int col = cb + wave * 16 + lm;
  const float bv = bias ? bias[col] : 0.0f;
#pragma unroll
  for (int v = 0; v < 8; ++v) {
    int m0 = v + 8 * half;
    float c0 = acc0[v] + bv;
    float c1 = acc1[v] + bv;
    if (mode == 1) { c0 = tanhf(c0); c1 = tanhf(c1); }
    C[(size_t)m0 * ncols + col]        = c0;
    C[(size_t)(m0 + 16) * ncols + col] = c1;
  }
}

// ---------------------------------------------------------------------------
// out[row] = W[row,:] . concat(x1[0:n1], x2[0:n2])      one wave per row
// ---------------------------------------------------------------------------
__global__ __launch_bounds__(256) void gemv_kernel(
    const float* __restrict__ W, int ldw,
    const float* __restrict__ x1, int n1,
    const float* __restrict__ x2, int n2,
    float* __restrict__ out, int nrows) {
  const int row = blockIdx.x * 8 + (threadIdx.x >> 5);
  if (row >= nrows) return;
  const int lane = threadIdx.x & 31;
  const float* wr = W + (size_t)row * ldw;
  float p = 0.0f;
  for (int k = lane * 4; k < n1; k += 128) {
    float4 w = *(const float4*)&wr[k];
    float4 x = *(const float4*)&x1[k];
    p += w.x * x.x + w.y * x.y + w.z * x.z + w.w * x.w;
  }
  if (x2) {
    for (int k = lane * 4; k < n2; k += 128) {
      float4 w = *(const float4*)&wr[n1 + k];
      float4 x = *(const float4*)&x2[k];
      p += w.x * x.x + w.y * x.y + w.z * x.z + w.w * x.w;
    }
  }
  p = waveReduceSum(p);
  if (lane == 0) out[row] = p;
}

// ---------------------------------------------------------------------------
// attn = softmax_n( ai[n,:] . W_attn )    single block, 8 waves x 4 rows
// ---------------------------------------------------------------------------
__global__ __launch_bounds__(256) void attn_softmax_kernel(
    const float* __restrict__ ai, const float* __restrict__ Wattn,
    float* __restrict__ attn) {
  __shared__ float lg[32];
  __shared__ float inv_sum;
  const int lane = threadIdx.x & 31;
  const int wave = threadIdx.x >> 5;
#pragma unroll
  for (int rr = 0; rr < 4; ++rr) {
    const int n = wave * 4 + rr;
    const float* r = ai + (size_t)n * 2048;
    float p = 0.0f;
    for (int k = lane * 4; k < 2048; k += 128) {
      float4 a = *(const float4*)&r[k];
      float4 w = *(const float4*)&Wattn[k];
      p += a.x * w.x + a.y * w.y + a.z * w.z + a.w * w.w;
    }
    p = waveReduceSum(p);
    if (lane == 0) lg[n] = p;
  }
  __syncthreads();
  if (threadIdx.x == 0) {
    float mx = lg[0];
    for (int n = 1; n < 32; ++n) mx = fmaxf(mx, lg[n]);
    float s = 0.0f;
    for (int n = 0; n < 32; ++n) { float e = expf(lg[n] - mx); lg[n] = e; s += e; }
    inv_sum = 1.0f / s;
  }
  __syncthreads();
  if (threadIdx.x < 32) attn[threadIdx.x] = lg[threadIdx.x] * inv_sum;
}

// merged[h] = sum_n attn[n] * hiddens[n,h]
__global__ __launch_bounds__(256) void merged_kernel(
    const float* __restrict__ hiddens, const float* __restrict__ attn,
    float* __restrict__ merged) {
  const int h = blockIdx.x * 256 + threadIdx.x;
  float a = 0.0f;
#pragma unroll
  for (int n = 0; n < 32; ++n) a += attn[n] * hiddens[(size_t)n * 2048 + h];
  merged[h] = a;
}

// mean / rstd of x[0:n]  -> out2 = {mean, 1/sqrt(var+eps)}
__global__ __launch_bounds__(256) void stats_kernel(
    const float* __restrict__ x, int n, float* __restrict__ out2) {
  __shared__ float s1[256], s2[256];
  float a = 0.0f, q = 0.0f;
  for (int i = threadIdx.x; i < n; i += 256) { float v = x[i]; a += v; q += v * v; }
  s1[threadIdx.x] = a; s2[threadIdx.x] = q;
  __syncthreads();
  for (int o = 128; o > 0; o >>= 1) {
    if (threadIdx.x < o) { s1[threadIdx.x] += s1[threadIdx.x + o]; s2[threadIdx.x] += s2[threadIdx.x + o]; }
    __syncthreads();
  }
  if (threadIdx.x == 0) {
    float m = s1[0] / n;
    float var = s2[0] / n - m * m;
    out2[0] = m;
    out2[1] = rsqrtf(var + 1e-5f);
  }
}

// out[h] = act( (x[h]-mean)*rstd*g[h] + b[h] )   mode 1: tanh
__global__ __launch_bounds__(256) void ln_act_kernel(
    const float* __restrict__ x, const float* __restrict__ st,
    const float* __restrict__ g, const float* __restrict__ b,
    float* __restrict__ out, int mode) {
  const int h = blockIdx.x * 256 + threadIdx.x;
  float v = (x[h] - st[0]) * st[1] * g[h] + b[h];
  if (mode == 1) v = tanhf(v);
  out[h] = v;
}

// per-row LN + sigmoid of f_lin, fused multiply by cells: fc[n,h]=f[n,h]*c[n,h]
__global__ __launch_bounds__(256) void f_row_kernel(
    const float* __restrict__ flin, const float* __restrict__ g,
    const float* __restrict__ b, const float* __restrict__ cells,
    float* __restrict__ fc) {
  __shared__ float s1[256], s2[256];
  __shared__ float mr[2];
  const int n = blockIdx.x;
  const float* x = flin + (size_t)n * 2048;
  float a = 0.0f, q = 0.0f;
  for (int i = threadIdx.x; i < 2048; i += 256) { float v = x[i]; a += v; q += v * v; }
  s1[threadIdx.x] = a; s2[threadIdx.x] = q;
  __syncthreads();
  for (int o = 128; o > 0; o >>= 1) {
    if (threadIdx.x < o) { s1[threadIdx.x] += s1[threadIdx.x + o]; s2[threadIdx.x] += s2[threadIdx.x + o]; }
    __syncthreads();
  }
  if (threadIdx.x == 0) {
    float m = s1[0] / 2048.0f;
    mr[0] = m;
    mr[1] = rsqrtf(s2[0] / 2048.0f - m * m + 1e-5f);
  }
  __syncthreads();
  for (int i = threadIdx.x; i < 2048; i += 256) {
    float v = (x[i] - mr[0]) * mr[1] * g[i] + b[i];
    fc[(size_t)n * 2048 + i] = sigmoidf_(v) * cells[(size_t)n * 2048 + i];
  }
}

// i/o/u gates + cell preactivation:  cellpre = i*u + sum_n fc[n,:]
__global__ __launch_bounds__(256) void iou_cell_kernel(
    const float* __restrict__ iou, const float* __restrict__ st,  // st[2..7]
    const float* __restrict__ gi, const float* __restrict__ bi,
    const float* __restrict__ go, const float* __restrict__ bo,
    const float* __restrict__ gu, const float* __restrict__ bu,
    const float* __restrict__ fc, float* __restrict__ cellpre,
    float* __restrict__ ovec) {
  const int h = blockIdx.x * 256 + threadIdx.x;
  float iv = sigmoidf_((iou[h]        - st[2]) * st[3] * gi[h] + bi[h]);
  float ov = sigmoidf_((iou[2048 + h] - st[4]) * st[5] * go[h] + bo[h]);
  float uv = tanhf(    (iou[4096 + h] - st[6]) * st[7] * gu[h] + bu[h]);
  float s = 0.0f;
#pragma unroll
  for (int n = 0; n < 32; ++n) s += fc[(size_t)n * 2048 + h];
  cellpre[h] = iv * uv + s;
  ovec[h] = ov;
}

// new_cell = LN(cellpre), new_h = o * tanh(new_cell);  out = [new_h | new_cell]
__global__ __launch_bounds__(256) void final_kernel(
    const float* __restrict__ cellpre, const float* __restrict__ st,
    const float* __restrict__ gc, const float* __restrict__ bc,
    const float* __restrict__ ovec, float* __restrict__ out) {
  const int h = blockIdx.x * 256 + threadIdx.x;
  float nc = (cellpre[h] - st[0]) * st[1] * gc[h] + bc[h];
  out[h] = ovec[h] * tanhf(nc);
  out[2048 + h] = nc;
}

// ---------------------------------------------------------------------------
extern "C" void kernel_launch(void* const* d_in, const int* in_sizes, int n_in,
                              void* d_out, int out_size, void* d_ws, size_t ws_size,
                              hipStream_t stream) {
  const float* input    = (const float*)d_in[0];
  const float* hiddens  = (const float*)d_in[1];
  const float* cells    = (const float*)d_in[2];
  const float* external = (const float*)d_in[3];
  const float* W_ai     = (const float*)d_in[4];
  const float* W_attn   = (const float*)d_in[5];
  const float* W_merge  = (const float*)d_in[6];
  const float* W_iou    = (const float*)d_in[7];
  const float* W_fi     = (const float*)d_in[8];
  const float* W_fh     = (const float*)d_in[9];
  const float* g_merge  = (const float*)d_in[10];
  const float* b_merge  = (const float*)d_in[11];
  const float* g_f      = (const float*)d_in[12];
  const float* b_f      = (const float*)d_in[13];
  const float* g_i      = (const float*)d_in[14];
  const float* b_i      = (const float*)d_in[15];
  const float* g_o      = (const float*)d_in[16];
  const float* b_o      = (const float*)d_in[17];
  const float* g_u      = (const float*)d_in[18];
  const float* b_u      = (const float*)d_in[19];
  const float* g_c      = (const float*)d_in[20];
  const float* b_c      = (const float*)d_in[21];

  float* ws        = (float*)d_ws;
  float* ws_ai     = ws;                 // [32,2048]
  float* ws_flin   = ws + 65536;         // [32,2048]
  float* ws_fi     = ws + 131072;        // [2048]
  float* ws_attn   = ws + 133120;        // [32]
  float* ws_merged = ws + 133152;        // [2048]
  float* ws_mpre   = ws + 135200;        // [2048]
  float* ws_mh     = ws + 137248;        // [2048]
  float* ws_iou    = ws + 139296;        // [6144]
  float* ws_fc     = ws + 145440;        // [32,2048]
  float* ws_cpre   = ws + 210976;        // [2048]
  float* ws_o      = ws + 213024;        // [2048]
  float* ws_st     = ws + 215072;        // [10] stats: merge,i,o,u,c

  dim3 b256(256);

  // ai = tanh([hiddens|external] @ W_ai^T)            (WMMA+TDM GEMM, K=4096)
  gemm32_wmma_kernel<<<dim3(16), b256, 0, stream>>>(
      hiddens, 2048, external, 2048, W_ai, nullptr, ws_ai, 2048, 1);
  // fi_vec = input @ W_fi^T                           (GEMV)
  gemv_kernel<<<dim3(256), b256, 0, stream>>>(
      W_fi, 2048, input, 2048, nullptr, 0, ws_fi, 2048);
  // f_lin = hiddens @ W_fh^T + fi_vec                 (WMMA+TDM GEMM, K=2048)
  gemm32_wmma_kernel<<<dim3(16), b256, 0, stream>>>(
      hiddens, 2048, nullptr, 0, W_fh, ws_fi, ws_flin, 2048, 0);

  attn_softmax_kernel<<<dim3(1), b256, 0, stream>>>(ws_ai, W_attn, ws_attn);
  merged_kernel<<<dim3(8), b256, 0, stream>>>(hiddens, ws_attn, ws_merged);

  gemv_kernel<<<dim3(256), b256, 0, stream>>>(
      W_merge, 2048, ws_merged, 2048, nullptr, 0, ws_mpre, 2048);
  stats_kernel<<<dim3(1), b256, 0, stream>>>(ws_mpre, 2048, ws_st + 0);
  ln_act_kernel<<<dim3(8), b256, 0, stream>>>(ws_mpre, ws_st + 0, g_merge, b_merge, ws_mh, 1);

  // iou = [input | merge_hidden] @ W_iou^T            (GEMV, 6144 rows, K=4096)
  gemv_kernel<<<dim3(768), b256, 0, stream>>>(
      W_iou, 4096, input, 2048, ws_mh, 2048, ws_iou, 6144);
  stats_kernel<<<dim3(1), b256, 0, stream>>>(ws_iou,        2048, ws_st + 2);
  stats_kernel<<<dim3(1), b256, 0, stream>>>(ws_iou + 2048, 2048, ws_st + 4);
  stats_kernel<<<dim3(1), b256, 0, stream>>>(ws_iou + 4096, 2048, ws_st + 6);

  f_row_kernel<<<dim3(32), b256, 0, stream>>>(ws_flin, g_f, b_f, cells, ws_fc);
  iou_cell_kernel<<<dim3(8), b256, 0, stream>>>(
      ws_iou, ws_st, g_i, b_i, g_o, b_o, g_u, b_u, ws_fc, ws_cpre, ws_o);
  stats_kernel<<<dim3(1), b256, 0, stream>>>(ws_cpre, 2048, ws_st + 8);
  final_kernel<<<dim3(8), b256, 0, stream>>>(
      ws_cpre, ws_st + 8, g_c, b_c, ws_o, (float*)d_out);
}